// GPSLayer_81037442940936
// MI455X (gfx1250) — compile-verified
//
#include <hip/hip_runtime.h>
#include <math.h>

typedef float v2f __attribute__((ext_vector_type(2)));
typedef float v8f __attribute__((ext_vector_type(8)));

#define WAVE 32
#define DD 128
#define NEG_BIG -3.0e38f

// ---------------------------------------------------------------- utilities
__device__ __forceinline__ float wave_sum(float x) {
  x += __shfl_xor(x, 16);
  x += __shfl_xor(x, 8);
  x += __shfl_xor(x, 4);
  x += __shfl_xor(x, 2);
  x += __shfl_xor(x, 1);
  return x;
}

__global__ void fill_kernel(float* __restrict__ p, float v, long long n) {
  long long i = (long long)blockIdx.x * blockDim.x + threadIdx.x;
  long long stride = (long long)gridDim.x * blockDim.x;
  for (; i < n; i += stride) p[i] = v;
}

// ---------------------------------------------------------------- degree / norm
__global__ void degree_kernel(const long long* __restrict__ ei,
                              float* __restrict__ deg, int E) {
  int i = blockIdx.x * blockDim.x + threadIdx.x;
  int stride = gridDim.x * blockDim.x;
  for (; i < E; i += stride) {
    int dst = (int)ei[(long long)E + i];
    atomicAdd(&deg[dst], 1.0f);
  }
}

__global__ void dinv_kernel(const float* __restrict__ deg,
                            float* __restrict__ dinv, int N) {
  int i = blockIdx.x * blockDim.x + threadIdx.x;
  if (i < N) {
    float d = deg[i];
    dinv[i] = (d > 0.0f) ? rsqrtf(d) : 0.0f;
  }
}

// ---------------------------------------------------------------- WMMA fp32 GEMM
// Out[N,Dout] = (relu?)(X[N,Din] @ W[Din,Dout] + bias)
// Block: 256 thr = 8 waves; block owns a 16-row strip of X (LDS-staged),
// each wave owns 16-column tiles. Uses V_WMMA_F32_16X16X4_F32, K-step 4.
__global__ void gemm_wmma_kernel(const float* __restrict__ X,
                                 const float* __restrict__ W,
                                 const float* __restrict__ bias,
                                 float* __restrict__ Out,
                                 int N, int Din, int Dout, int relu) {
  __shared__ float As[16 * 260];           // supports Din<=256, pad +4 (bank-safe)
  const int ldst = Din + 4;
  const int m0 = blockIdx.x * 16;
  const int tid = threadIdx.x;

  for (int idx = tid; idx < 16 * Din; idx += blockDim.x) {
    int r = idx / Din, c = idx - r * Din;
    int row = m0 + r;
    As[r * ldst + c] = (row < N) ? X[(long long)row * Din + c] : 0.0f;
  }
  __syncthreads();

  const int wave = tid / WAVE;
  const int lane = tid % WAVE;
  const int half = lane >> 4;              // 0: lanes 0-15, 1: lanes 16-31
  const int lp   = lane & 15;

  const int ntiles = Dout >> 4;
  for (int t = wave; t < ntiles; t += 8) {
    const int n0 = t * 16;
    v8f acc = {};
#pragma unroll 4
    for (int k = 0; k < Din; k += 4) {
      const int ka = k + 2 * half;         // A/B 32-bit layout: K = 2*half + vgpr
      v2f a, b;
      a.x = As[lp * ldst + ka];
      a.y = As[lp * ldst + ka + 1];
      b.x = W[(long long)ka * Dout + n0 + lp];
      b.y = W[(long long)(ka + 1) * Dout + n0 + lp];
      acc = __builtin_amdgcn_wmma_f32_16x16x4_f32(
          false, a, false, b, (short)0, acc, false, false);
    }
    const float bv = bias ? bias[n0 + lp] : 0.0f;
#pragma unroll
    for (int r = 0; r < 8; ++r) {          // C/D: row = m0 + r + 8*half, col = n0+lp
      int row = m0 + r + 8 * half;
      if (row < N) {
        float v = acc[r] + bv;
        if (relu) v = fmaxf(v, 0.0f);
        Out[(long long)row * Dout + n0 + lp] = v;
      }
    }
  }
}

// ---------------------------------------------------------------- GCN scatter
// local[dst] += h[src] * dinv[src]*dinv[dst]   (wave per edge, 4 floats/lane)
__global__ void gcn_scatter_kernel(const long long* __restrict__ ei,
                                   const float* __restrict__ h,
                                   const float* __restrict__ dinv,
                                   float* __restrict__ outp, int E) {
  long long gt = (long long)blockIdx.x * blockDim.x + threadIdx.x;
  long long nwaves = ((long long)gridDim.x * blockDim.x) / WAVE;
  long long wid = gt / WAVE;
  int lane = (int)(gt % WAVE);
  for (long long e = wid; e < E; e += nwaves) {
    int src = (int)ei[e];
    int dst = (int)ei[(long long)E + e];
    float norm = dinv[src] * dinv[dst];
    float4 hv = reinterpret_cast<const float4*>(h + (long long)src * DD)[lane];
    float* op = outp + (long long)dst * DD + lane * 4;
    atomicAdd(op + 0, hv.x * norm);
    atomicAdd(op + 1, hv.y * norm);
    atomicAdd(op + 2, hv.z * norm);
    atomicAdd(op + 3, hv.w * norm);
  }
}

// ---------------------------------------------------------------- attention scores
// scores[e][h] = dot(q[dst][h], k[src][h]) / sqrt(32); segment-max into mbuf
__global__ void attn_score_kernel(const long long* __restrict__ ei,
                                  const float* __restrict__ q,
                                  const float* __restrict__ k,
                                  float* __restrict__ sc,
                                  float* __restrict__ mbuf, int E) {
  long long gt = (long long)blockIdx.x * blockDim.x + threadIdx.x;
  long long nwaves = ((long long)gridDim.x * blockDim.x) / WAVE;
  long long wid = gt / WAVE;
  int lane = (int)(gt % WAVE);
  const float scale = 0.17677669529663687f;     // 1/sqrt(32)
  for (long long e = wid; e < E; e += nwaves) {
    int src = (int)ei[e];
    int dst = (int)ei[(long long)E + e];
    float4 qv = reinterpret_cast<const float4*>(q + (long long)dst * DD)[lane];
    float4 kv = reinterpret_cast<const float4*>(k + (long long)src * DD)[lane];
    float p = qv.x * kv.x + qv.y * kv.y + qv.z * kv.z + qv.w * kv.w;
    // reduce within each 8-lane head group (head = lane>>3, HD=32 = 8 lanes * 4)
    p += __shfl_xor(p, 4);
    p += __shfl_xor(p, 2);
    p += __shfl_xor(p, 1);
    if ((lane & 7) == 0) {
      int hh = lane >> 3;
      float s = p * scale;
      sc[e * 4 + hh] = s;
      __hip_atomic_fetch_max(&mbuf[(long long)dst * 4 + hh], s,
                             __ATOMIC_RELAXED, __HIP_MEMORY_SCOPE_AGENT);
    }
  }
}

// ---------------------------------------------------------------- exp + denom
__global__ void attn_exp_kernel(const long long* __restrict__ ei,
                                float* __restrict__ sc,
                                const float* __restrict__ mbuf,
                                float* __restrict__ denom, int E) {
  int i = blockIdx.x * blockDim.x + threadIdx.x;
  int stride = gridDim.x * blockDim.x;
  for (int e = i; e < E; e += stride) {
    int dst = (int)ei[(long long)E + e];
#pragma unroll
    for (int hh = 0; hh < 4; ++hh) {
      float s = sc[(long long)e * 4 + hh];
      float ex = expf(s - mbuf[(long long)dst * 4 + hh]);
      sc[(long long)e * 4 + hh] = ex;
      atomicAdd(&denom[(long long)dst * 4 + hh], ex);
    }
  }
}

// ---------------------------------------------------------------- weighted aggregate
// agg[dst] += v[src] * (e / denom[dst])   (wave per edge)
__global__ void attn_agg_kernel(const long long* __restrict__ ei,
                                const float* __restrict__ vv,
                                const float* __restrict__ sc,
                                const float* __restrict__ denom,
                                float* __restrict__ agg, int E) {
  long long gt = (long long)blockIdx.x * blockDim.x + threadIdx.x;
  long long nwaves = ((long long)gridDim.x * blockDim.x) / WAVE;
  long long wid = gt / WAVE;
  int lane = (int)(gt % WAVE);
  int hh = lane >> 3;
  for (long long e = wid; e < E; e += nwaves) {
    int src = (int)ei[e];
    int dst = (int)ei[(long long)E + e];
    float alpha = sc[e * 4 + hh] / denom[(long long)dst * 4 + hh];
    float4 val = reinterpret_cast<const float4*>(vv + (long long)src * DD)[lane];
    float* op = agg + (long long)dst * DD + lane * 4;
    atomicAdd(op + 0, val.x * alpha);
    atomicAdd(op + 1, val.y * alpha);
    atomicAdd(op + 2, val.z * alpha);
    atomicAdd(op + 3, val.w * alpha);
  }
}

// ---------------------------------------------------------------- combine + LN1
// beta gate, xc = lw*local + gw*global, xn = LN(2*xc)   (wave per node)
__global__ void combine_kernel(const float* __restrict__ agg,
                               const float* __restrict__ xr,
                               const float* __restrict__ loc,
                               const float* __restrict__ b_gcn,
                               const float* __restrict__ w_beta,
                               const float* __restrict__ ln1g,
                               const float* __restrict__ ln1b,
                               const float* __restrict__ lwp,
                               const float* __restrict__ gwp,
                               float* __restrict__ xn, int N) {
  long long gt = (long long)blockIdx.x * blockDim.x + threadIdx.x;
  long long nwaves = ((long long)gridDim.x * blockDim.x) / WAVE;
  long long wid = gt / WAVE;
  int lane = (int)(gt % WAVE);
  const float lw = lwp[0], gw = gwp[0];
  for (long long n = wid; n < N; n += nwaves) {
    float4 a  = reinterpret_cast<const float4*>(agg + n * DD)[lane];
    float4 r  = reinterpret_cast<const float4*>(xr + n * DD)[lane];
    float4 l  = reinterpret_cast<const float4*>(loc + n * DD)[lane];
    float4 bg = reinterpret_cast<const float4*>(b_gcn)[lane];
    l.x += bg.x; l.y += bg.y; l.z += bg.z; l.w += bg.w;
    float4 w0 = reinterpret_cast<const float4*>(w_beta)[lane];
    float4 w1 = reinterpret_cast<const float4*>(w_beta + DD)[lane];
    float4 w2 = reinterpret_cast<const float4*>(w_beta + 2 * DD)[lane];
    float dp = a.x * w0.x + a.y * w0.y + a.z * w0.z + a.w * w0.w
             + r.x * w1.x + r.y * w1.y + r.z * w1.z + r.w * w1.w
             + (a.x - r.x) * w2.x + (a.y - r.y) * w2.y
             + (a.z - r.z) * w2.z + (a.w - r.w) * w2.w;
    dp = wave_sum(dp);
    float beta = 1.0f / (1.0f + expf(-dp));
    float4 y;
    y.x = 2.0f * (lw * l.x + gw * (beta * r.x + (1.0f - beta) * a.x));
    y.y = 2.0f * (lw * l.y + gw * (beta * r.y + (1.0f - beta) * a.y));
    y.z = 2.0f * (lw * l.z + gw * (beta * r.z + (1.0f - beta) * a.z));
    y.w = 2.0f * (lw * l.w + gw * (beta * r.w + (1.0f - beta) * a.w));
    float s1 = wave_sum(y.x + y.y + y.z + y.w);
    float s2 = wave_sum(y.x * y.x + y.y * y.y + y.z * y.z + y.w * y.w);
    float mean = s1 * (1.0f / DD);
    float var  = s2 * (1.0f / DD) - mean * mean;
    float inv  = rsqrtf(var + 1e-5f);
    float4 g = reinterpret_cast<const float4*>(ln1g)[lane];
    float4 b = reinterpret_cast<const float4*>(ln1b)[lane];
    float4 o;
    o.x = (y.x - mean) * inv * g.x + b.x;
    o.y = (y.y - mean) * inv * g.y + b.y;
    o.z = (y.z - mean) * inv * g.z + b.z;
    o.w = (y.w - mean) * inv * g.w + b.w;
    reinterpret_cast<float4*>(xn + n * DD)[lane] = o;
  }
}

// ---------------------------------------------------------------- final residual + LN2
__global__ void final_ln_kernel(const float* __restrict__ ff2,
                                const float* __restrict__ xn,
                                const float* __restrict__ ln2g,
                                const float* __restrict__ ln2b,
                                float* __restrict__ out, int N) {
  long long gt = (long long)blockIdx.x * blockDim.x + threadIdx.x;
  long long nwaves = ((long long)gridDim.x * blockDim.x) / WAVE;
  long long wid = gt / WAVE;
  int lane = (int)(gt % WAVE);
  for (long long n = wid; n < N; n += nwaves) {
    float4 f = reinterpret_cast<const float4*>(ff2 + n * DD)[lane];
    float4 x = reinterpret_cast<const float4*>(xn + n * DD)[lane];
    float4 y;
    y.x = f.x + x.x; y.y = f.y + x.y; y.z = f.z + x.z; y.w = f.w + x.w;
    float s1 = wave_sum(y.x + y.y + y.z + y.w);
    float s2 = wave_sum(y.x * y.x + y.y * y.y + y.z * y.z + y.w * y.w);
    float mean = s1 * (1.0f / DD);
    float var  = s2 * (1.0f / DD) - mean * mean;
    float inv  = rsqrtf(var + 1e-5f);
    float4 g = reinterpret_cast<const float4*>(ln2g)[lane];
    float4 b = reinterpret_cast<const float4*>(ln2b)[lane];
    float4 o;
    o.x = (y.x - mean) * inv * g.x + b.x;
    o.y = (y.y - mean) * inv * g.y + b.y;
    o.z = (y.z - mean) * inv * g.z + b.z;
    o.w = (y.w - mean) * inv * g.w + b.w;
    reinterpret_cast<float4*>(out + n * DD)[lane] = o;
  }
}

// ---------------------------------------------------------------- launch
extern "C" void kernel_launch(void* const* d_in, const int* in_sizes, int n_in,
                              void* d_out, int out_size, void* d_ws, size_t ws_size,
                              hipStream_t stream) {
  const float* x       = (const float*)d_in[0];
  const long long* ei  = (const long long*)d_in[1];   // int64 edge_index [2,E]
  const float* w_gcn   = (const float*)d_in[2];
  const float* b_gcn   = (const float*)d_in[3];
  const float* w_q     = (const float*)d_in[4];
  const float* b_q     = (const float*)d_in[5];
  const float* w_k     = (const float*)d_in[6];
  const float* b_k     = (const float*)d_in[7];
  const float* w_v     = (const float*)d_in[8];
  const float* b_v     = (const float*)d_in[9];
  const float* w_skip  = (const float*)d_in[10];
  const float* b_skip  = (const float*)d_in[11];
  const float* w_beta  = (const float*)d_in[12];
  const float* ln1_g   = (const float*)d_in[13];
  const float* ln1_b   = (const float*)d_in[14];
  const float* ln2_g   = (const float*)d_in[15];
  const float* ln2_b   = (const float*)d_in[16];
  const float* w1      = (const float*)d_in[17];
  const float* b1      = (const float*)d_in[18];
  const float* w2      = (const float*)d_in[19];
  const float* b2      = (const float*)d_in[20];
  const float* lw      = (const float*)d_in[21];
  const float* gw      = (const float*)d_in[22];

  const int N = in_sizes[0] / DD;
  const int E = in_sizes[1] / 2;
  const long long ND = (long long)N * DD;

  float* ws    = (float*)d_ws;
  float* h     = ws;
  float* q     = ws + 1 * ND;
  float* k     = ws + 2 * ND;
  float* v     = ws + 3 * ND;
  float* skip  = ws + 4 * ND;
  float* loc   = ws + 5 * ND;
  float* agg   = ws + 6 * ND;
  float* xn    = ws + 7 * ND;
  float* ff1   = ws + 8 * ND;               // N x 256
  float* ff2   = ws + 10 * ND;
  float* deg   = ws + 11 * ND;
  float* dinv  = deg + N;
  float* mbuf  = dinv + N;                  // N x 4
  float* denom = mbuf + 4LL * N;            // N x 4
  float* sc    = denom + 4LL * N;           // E x 4

  float* out = (float*)d_out;

  // re-init accumulators (graph replays reuse the workspace)
  fill_kernel<<<2048, 256, 0, stream>>>(deg, 0.0f, (long long)N);
  fill_kernel<<<2048, 256, 0, stream>>>(loc, 0.0f, ND);
  fill_kernel<<<2048, 256, 0, stream>>>(agg, 0.0f, ND);
  fill_kernel<<<2048, 256, 0, stream>>>(mbuf, NEG_BIG, 4LL * N);
  fill_kernel<<<2048, 256, 0, stream>>>(denom, 0.0f, 4LL * N);

  // degrees and symmetric norm
  degree_kernel<<<4096, 256, 0, stream>>>(ei, deg, E);
  dinv_kernel<<<(N + 255) / 256, 256, 0, stream>>>(deg, dinv, N);

  // dense projections (fp32 WMMA)
  const int gblocks = (N + 15) / 16;
  gemm_wmma_kernel<<<gblocks, 256, 0, stream>>>(x, w_gcn, nullptr, h,    N, DD, DD, 0);
  gemm_wmma_kernel<<<gblocks, 256, 0, stream>>>(x, w_q,   b_q,    q,    N, DD, DD, 0);
  gemm_wmma_kernel<<<gblocks, 256, 0, stream>>>(x, w_k,   b_k,    k,    N, DD, DD, 0);
  gemm_wmma_kernel<<<gblocks, 256, 0, stream>>>(x, w_v,   b_v,    v,    N, DD, DD, 0);
  gemm_wmma_kernel<<<gblocks, 256, 0, stream>>>(x, w_skip, b_skip, skip, N, DD, DD, 0);

  // GCN message passing
  gcn_scatter_kernel<<<8192, 256, 0, stream>>>(ei, h, dinv, loc, E);

  // graph attention
  attn_score_kernel<<<8192, 256, 0, stream>>>(ei, q, k, sc, mbuf, E);
  attn_exp_kernel<<<4096, 256, 0, stream>>>(ei, sc, mbuf, denom, E);
  attn_agg_kernel<<<8192, 256, 0, stream>>>(ei, v, sc, denom, agg, E);

  // gate + combine + LN1
  combine_kernel<<<8192, 256, 0, stream>>>(agg, skip, loc, b_gcn, w_beta,
                                           ln1_g, ln1_b, lw, gw, xn, N);

  // FFN (relu fused into first GEMM) + final LN
  gemm_wmma_kernel<<<gblocks, 256, 0, stream>>>(xn, w1, b1, ff1, N, DD, 2 * DD, 1);
  gemm_wmma_kernel<<<gblocks, 256, 0, stream>>>(ff1, w2, b2, ff2, N, 2 * DD, DD, 0);
  final_ln_kernel<<<8192, 256, 0, stream>>>(ff2, xn, ln2_g, ln2_b, out, N);
}